// GeneralLayer_88519275970680
// MI455X (gfx1250) — compile-verified
//
#include <hip/hip_runtime.h>
#include <math.h>

#define NN     50000
#define EE     800000
#define DD     128
#define MM     3
#define SLOPE  0.2f

typedef __attribute__((ext_vector_type(2))) float v2f;
typedef __attribute__((ext_vector_type(8))) float v8f;

__device__ __forceinline__ void atomicMaxF(float* addr, float v) {
  // exact float max via monotonic integer mappings (init value is -inf)
  if (v >= 0.0f) atomicMax((int*)addr, __float_as_int(v));
  else           atomicMin((unsigned int*)addr, __float_as_uint(v));
}

// ---------------------------------------------------------------------------
// feat[N,128] = h[N,128] @ Wm[128,128], fp32 WMMA 16x16x4 (exact fp32 math).
// One wave per 16x16 output tile; block = 8 waves = one 16-row stripe.
// A layout: lane%16 = M row, VGPR j holds K = k0 + 2*(lane/16) + j
// B layout: lane%16 = N col, VGPR j holds K = k0 + 2*(lane/16) + j
// C/D layout: lane%16 = N col, VGPR j holds M = j + 8*(lane/16)
// ---------------------------------------------------------------------------
__global__ void gemm_feat(const float* __restrict__ h,
                          const float* __restrict__ Wm,
                          float* __restrict__ feat) {
  const int lane = threadIdx.x & 31;
  const int wave = threadIdx.x >> 5;        // column tile 0..7
  const int g    = lane >> 4;               // K-pair group
  const int r    = lane & 15;
  const int i0   = blockIdx.x << 4;         // 3125 row tiles (50000/16 exact)
  const int j0   = wave << 4;
  v8f acc = {0.f, 0.f, 0.f, 0.f, 0.f, 0.f, 0.f, 0.f};
  const float* arow = h  + (size_t)(i0 + r) * DD;
  const float* bcol = Wm + j0 + r;
  #pragma unroll 4
  for (int k0 = 0; k0 < DD; k0 += 4) {
    const int ka = k0 + 2 * g;
    v2f a, b;
    a.x = arow[ka];
    a.y = arow[ka + 1];
    b.x = bcol[(size_t)ka * DD];
    b.y = bcol[(size_t)(ka + 1) * DD];
    acc = __builtin_amdgcn_wmma_f32_16x16x4_f32(false, a, false, b,
                                                (short)0, acc, false, false);
  }
  float* drow = feat + (size_t)(i0 + 8 * g) * DD + j0 + r;
  #pragma unroll
  for (int jj = 0; jj < 8; ++jj)
    drow[(size_t)jj * DD] = acc[jj];
}

// el[n] = feat[n]·al, er[n] = feat[n]·ar  (one wave per node, coalesced 512B)
__global__ void node_elr(const float* __restrict__ feat,
                         const float* __restrict__ al,
                         const float* __restrict__ ar,
                         float* __restrict__ el, float* __restrict__ er) {
  const int lane = threadIdx.x & 31;
  const int node = (int)((blockIdx.x * blockDim.x + threadIdx.x) >> 5);
  if (node >= NN) return;
  const float4 f = *(const float4*)(feat + (size_t)node * DD + lane * 4);
  const float4 a = *(const float4*)(al + lane * 4);
  const float4 b = *(const float4*)(ar + lane * 4);
  float sl = f.x * a.x + f.y * a.y + f.z * a.z + f.w * a.w;
  float sr = f.x * b.x + f.y * b.y + f.z * b.z + f.w * b.w;
  #pragma unroll
  for (int off = 16; off; off >>= 1) {
    sl += __shfl_xor(sl, off, 32);
    sr += __shfl_xor(sr, off, 32);
  }
  if (lane == 0) { el[node] = sl; er[node] = sr; }
}

__global__ void init_neg_inf(float* __restrict__ p, int n) {
  int i = blockIdx.x * blockDim.x + threadIdx.x;
  if (i < n) p[i] = -__builtin_inff();
}

// e = leaky_relu(el[src]+er[dst]); segment-max over dst
__global__ void edge_e(const int* __restrict__ src, const int* __restrict__ dst,
                       const float* __restrict__ el, const float* __restrict__ er,
                       float* __restrict__ ebuf, float* __restrict__ dstmax) {
  int e = blockIdx.x * blockDim.x + threadIdx.x;
  if (e >= EE) return;
  float x = el[src[e]] + er[dst[e]];
  float v = x > 0.0f ? x : SLOPE * x;
  ebuf[e] = v;
  atomicMaxF(dstmax + dst[e], v);
}

// w = exp(e - max[dst]); denom[dst] += w
__global__ void edge_w(const int* __restrict__ dst,
                       const float* __restrict__ dstmax,
                       float* __restrict__ ebuf, float* __restrict__ denom) {
  int e = blockIdx.x * blockDim.x + threadIdx.x;
  if (e >= EE) return;
  int d = dst[e];
  float w = expf(ebuf[e] - dstmax[d]);
  ebuf[e] = w;
  atomicAdd(denom + d, w);
}

// z[dst] += alpha * feat[src]; 32 lanes per edge, 4 features per lane
__global__ void edge_aggr(const int* __restrict__ src, const int* __restrict__ dst,
                          const float* __restrict__ ebuf,
                          const float* __restrict__ denom,
                          const float* __restrict__ feat,
                          float* __restrict__ zm) {
  int gid  = blockIdx.x * blockDim.x + threadIdx.x;   // < EE*32 = 25.6M
  int e    = gid >> 5;
  int lane = threadIdx.x & 31;
  if (e >= EE) return;
  int s = src[e], d = dst[e];
  float alpha = ebuf[e] / fmaxf(denom[d], 1e-16f);
  const float4 f = *(const float4*)(feat + (size_t)s * DD + lane * 4);
  float* zp = zm + (size_t)d * DD + lane * 4;
  atomicAdd(zp + 0, alpha * f.x);
  atomicAdd(zp + 1, alpha * f.y);
  atomicAdd(zp + 2, alpha * f.z);
  atomicAdd(zp + 3, alpha * f.w);
}

// score[n,m] = tanh(z[n,m] @ W1 + b1) · w2 ; accumulate sum over n per m
__global__ void sem_score(const float* __restrict__ z,
                          const float* __restrict__ W1,
                          const float* __restrict__ b1,
                          const float* __restrict__ w2,
                          float* __restrict__ scoresum) {
  __shared__ float zs[DD];
  __shared__ float red[DD];
  const int j = threadIdx.x;
  const int n = blockIdx.x;
  const int m = blockIdx.y;
  const float* zrow = z + ((size_t)m * NN + n) * DD;
  zs[j] = zrow[j];
  __syncthreads();
  float acc = b1[j];
  #pragma unroll 8
  for (int d = 0; d < DD; ++d)
    acc = fmaf(zs[d], W1[(size_t)d * DD + j], acc);
  red[j] = tanhf(acc) * w2[j];
  __syncthreads();
  for (int s = 64; s > 0; s >>= 1) {
    if (j < s) red[j] += red[j + s];
    __syncthreads();
  }
  if (j == 0) atomicAdd(scoresum + m, red[0]);
}

__global__ void beta_kernel(const float* __restrict__ scoresum,
                            float* __restrict__ beta) {
  float s0 = scoresum[0] * (1.0f / NN);
  float s1 = scoresum[1] * (1.0f / NN);
  float s2 = scoresum[2] * (1.0f / NN);
  float mx = fmaxf(s0, fmaxf(s1, s2));
  float e0 = expf(s0 - mx), e1 = expf(s1 - mx), e2 = expf(s2 - mx);
  float inv = 1.0f / (e0 + e1 + e2);
  beta[0] = e0 * inv; beta[1] = e1 * inv; beta[2] = e2 * inv;
}

__global__ void combine(const float* __restrict__ z,
                        const float* __restrict__ beta,
                        float* __restrict__ out) {
  int i = blockIdx.x * blockDim.x + threadIdx.x;
  if (i >= NN * DD) return;
  const size_t stride = (size_t)NN * DD;
  out[i] = beta[0] * z[i] + beta[1] * z[i + stride] + beta[2] * z[i + 2 * stride];
}

// per-feature sum and sum-of-squares for BN batch statistics
__global__ void bn_stats(const float* __restrict__ out,
                         float* __restrict__ bnsum, float* __restrict__ bnsq) {
  const int d = threadIdx.x;                 // 128 threads: one feature each
  float s = 0.0f, q = 0.0f;
  for (int n = blockIdx.x; n < NN; n += gridDim.x) {
    float v = out[(size_t)n * DD + d];
    s += v; q += v * v;
  }
  atomicAdd(bnsum + d, s);
  atomicAdd(bnsq + d, q);
}

// BN (batch stats) -> ReLU -> row-wise L2 normalize, in place in d_out
__global__ void bn_relu_norm(float* __restrict__ out,
                             const float* __restrict__ bnsum,
                             const float* __restrict__ bnsq,
                             const float* __restrict__ gamma,
                             const float* __restrict__ betaBN) {
  __shared__ float red[DD];
  const int d = threadIdx.x;
  const int n = blockIdx.x;
  float mu  = bnsum[d] * (1.0f / NN);
  float var = bnsq[d] * (1.0f / NN) - mu * mu;
  float x = out[(size_t)n * DD + d];
  float y = (x - mu) * rsqrtf(var + 1e-5f) * gamma[d] + betaBN[d];
  y = fmaxf(y, 0.0f);
  red[d] = y * y;
  __syncthreads();
  for (int s = 64; s > 0; s >>= 1) {
    if (d < s) red[d] += red[d + s];
    __syncthreads();
  }
  float nrm = fmaxf(sqrtf(red[0]), 1e-12f);
  out[(size_t)n * DD + d] = y / nrm;
}

// ---------------------------------------------------------------------------

static inline size_t alignUp(size_t v, size_t a) { return (v + a - 1) & ~(a - 1); }

extern "C" void kernel_launch(void* const* d_in, const int* in_sizes, int n_in,
                              void* d_out, int out_size, void* d_ws, size_t ws_size,
                              hipStream_t stream) {
  const float* h        = (const float*)d_in[0];   // [N,128]
  const float* W        = (const float*)d_in[1];   // [3,128,128]
  const float* attn_l   = (const float*)d_in[2];   // [3,128]
  const float* attn_r   = (const float*)d_in[3];   // [3,128]
  const float* sem_W1   = (const float*)d_in[4];   // [128,128]
  const float* sem_b1   = (const float*)d_in[5];   // [128]
  const float* sem_w2   = (const float*)d_in[6];   // [128]
  const float* bn_gamma = (const float*)d_in[7];   // [128]
  const float* bn_beta  = (const float*)d_in[8];   // [128]
  const int*   src_idx  = (const int*)d_in[9];     // [3,E]
  const int*   dst_idx  = (const int*)d_in[10];    // [3,E]
  float* out = (float*)d_out;

  // workspace carve-out (~106 MB)
  char* p = (char*)d_ws;
  float* feat     = (float*)p; p += alignUp((size_t)NN * DD * 4, 256);
  float* z        = (float*)p; p += alignUp((size_t)MM * NN * DD * 4, 256);
  float* el       = (float*)p; p += alignUp((size_t)NN * 4, 256);
  float* er       = (float*)p; p += alignUp((size_t)NN * 4, 256);
  float* dstmax   = (float*)p; p += alignUp((size_t)NN * 4, 256);
  float* denom    = (float*)p; p += alignUp((size_t)NN * 4, 256);
  float* ebuf     = (float*)p; p += alignUp((size_t)EE * 4, 256);
  float* scoresum = (float*)p; p += alignUp((size_t)MM * 4, 256);
  float* betaBuf  = (float*)p; p += alignUp((size_t)MM * 4, 256);
  float* bnsum    = (float*)p; p += alignUp((size_t)DD * 4, 256);
  float* bnsq     = (float*)p; p += alignUp((size_t)DD * 4, 256);

  hipMemsetAsync(z,        0, (size_t)MM * NN * DD * 4, stream);
  hipMemsetAsync(scoresum, 0, (size_t)MM * 4, stream);
  hipMemsetAsync(bnsum,    0, (size_t)DD * 4, stream);
  hipMemsetAsync(bnsq,     0, (size_t)DD * 4, stream);

  for (int m = 0; m < MM; ++m) {
    gemm_feat<<<NN / 16, 256, 0, stream>>>(h, W + (size_t)m * DD * DD, feat);
    node_elr<<<(NN * 32 + 255) / 256, 256, 0, stream>>>(
        feat, attn_l + m * DD, attn_r + m * DD, el, er);
    init_neg_inf<<<(NN + 255) / 256, 256, 0, stream>>>(dstmax, NN);
    hipMemsetAsync(denom, 0, (size_t)NN * 4, stream);
    const int* sm = src_idx + (size_t)m * EE;
    const int* dm = dst_idx + (size_t)m * EE;
    edge_e<<<(EE + 255) / 256, 256, 0, stream>>>(sm, dm, el, er, ebuf, dstmax);
    edge_w<<<(EE + 255) / 256, 256, 0, stream>>>(dm, dstmax, ebuf, denom);
    edge_aggr<<<(EE * 32) / 256, 256, 0, stream>>>(
        sm, dm, ebuf, denom, feat, z + (size_t)m * NN * DD);
  }

  sem_score<<<dim3(NN, MM), DD, 0, stream>>>(z, sem_W1, sem_b1, sem_w2, scoresum);
  beta_kernel<<<1, 1, 0, stream>>>(scoresum, betaBuf);
  combine<<<(NN * DD + 255) / 256, 256, 0, stream>>>(z, betaBuf, out);
  bn_stats<<<256, DD, 0, stream>>>(out, bnsum, bnsq);
  bn_relu_norm<<<NN, DD, 0, stream>>>(out, bnsum, bnsq, bn_gamma, bn_beta);
}